// EKFModel_48687749268198
// MI455X (gfx1250) — compile-verified
//
#include <hip/hip_runtime.h>

// CDNA5 / gfx1250 wave32 EKF: one wave per batch element, 18x18 covariance
// chains via V_WMMA_F32_16X16X4_F32 (f32 tensor path, precision-preserving).
// All LDS matrices padded to 20x20 with rows/cols 18-19 kept zero so that
// WMMA fragment loads need no bounds guards (garbage can only reach D rows/
// cols >= 18, which are never stored).

typedef __attribute__((ext_vector_type(2))) float v2f;
typedef __attribute__((ext_vector_type(8))) float v8f;

#define WAVES   4
#define LDA     20      // padded row stride for 18x18 matrices
#define MSZ     400     // 20x20 floats
#define ELEM_F  2252    // floats of LDS per wave

// per-wave LDS offsets (floats, all 8-byte aligned)
#define OFF_P    0      // 20x20 covariance
#define OFF_PHI  400    // 20x20 Phi
#define OFF_F    800    // 20x20 F (reused as update scratch)
#define OFF_T1   1200   // temp (F@F, Phi@P)
#define OFF_T2   1600   // temp (pre-symmetrize P)
#define OFF_PG   2000   // 20x12 Phi@G (K=12 exact, no padding needed)
#define OFF_RS   2240   // 9 floats: R matrix for current step

__device__ __forceinline__ v8f wmma4(v2f a, v2f b, v8f c) {
  // D(16x16,f32) = A(16x4,f32) * B(4x16,f32) + C
  return __builtin_amdgcn_wmma_f32_16x16x4_f32(
      /*neg_a=*/false, a, /*neg_b=*/false, b,
      /*c_mod=*/(short)0, c, /*reuse_a=*/false, /*reuse_b=*/false);
}

// A fragment: 16x4 tile at (mi*16, k0), row-major stride sa. Unguarded:
// k0+2*lh .. +1 stay inside the (zero-padded) 20-col row; rows >= 18 only
// feed unstored D rows.
__device__ __forceinline__ v2f fragA(const float* A, int sa,
                                     int mi, int k0, int lr, int lh) {
  int m = mi * 16 + lr;
  int k = k0 + 2 * lh;
  v2f a;
  a.x = A[m * sa + k];
  a.y = A[m * sa + k + 1];
  return a;
}

// B fragment: 4x16 tile at (k0, ni*16) of KxN row-major matrix (stride sb).
__device__ __forceinline__ v2f fragB(const float* B, int sb,
                                     int ni, int k0, int lr, int lh) {
  int n = ni * 16 + lr;
  int k = k0 + 2 * lh;
  v2f b;
  b.x = B[ k      * sb + n];
  b.y = B[(k + 1) * sb + n];
  return b;
}

// B fragment where logical B[k][n] = Bt[n][k]; Bt row-major stride sbt.
__device__ __forceinline__ v2f fragBT(const float* Bt, int sbt,
                                      int ni, int k0, int lr, int lh) {
  int n = ni * 16 + lr;
  int k = k0 + 2 * lh;
  v2f b;
  b.x = Bt[n * sbt + k];
  b.y = Bt[n * sbt + k + 1];
  return b;
}

// Guarded C/D store (guards fold to compile-time / cheap lh,lr compares).
__device__ __forceinline__ void storeD(float* C, v8f acc,
                                       int mi, int ni, int lr, int lh) {
  #pragma unroll
  for (int v = 0; v < 8; ++v) {
    int m = mi * 16 + v + 8 * lh;
    int n = ni * 16 + lr;
    if (m < 18 && n < 18) C[m * LDA + n] = acc[v];
  }
}

// C(18x18) = A(18x18) @ B(18x18), padded stride LDA in LDS.
__device__ __forceinline__ void matmul18(const float* A, const float* B,
                                         float* C, int lr, int lh) {
  #pragma unroll
  for (int mi = 0; mi < 2; ++mi) {
    #pragma unroll
    for (int ni = 0; ni < 2; ++ni) {
      v8f acc = {};
      #pragma unroll
      for (int k0 = 0; k0 < 20; k0 += 4) {
        v2f a = fragA(A, LDA, mi, k0, lr, lh);
        v2f b = fragB(B, LDA, ni, k0, lr, lh);
        acc = wmma4(a, b, acc);
      }
      storeD(C, acc, mi, ni, lr, lh);
    }
  }
}

__device__ __forceinline__ float skew_ij(float x, float y, float z, int i, int j) {
  // [[0,-z,y],[z,0,-x],[-y,x,0]]
  if (i == 0) return (j == 0) ? 0.0f : ((j == 1) ? -z : y);
  if (i == 1) return (j == 0) ? z : ((j == 1) ? 0.0f : -x);
  return (j == 0) ? -y : ((j == 1) ? x : 0.0f);
}

__global__ __launch_bounds__(32 * WAVES)
void ekf_kernel(const float* __restrict__ dts, const float* __restrict__ wa,
                const float* __restrict__ Rg,  const float* __restrict__ vel,
                const float* __restrict__ grav,const float* __restrict__ H0g,
                const float* __restrict__ H1g, const float* __restrict__ prot,
                const float* __restrict__ ptrans, const float* __restrict__ vrot,
                const float* __restrict__ vtrans, const float* __restrict__ vrstd,
                const float* __restrict__ vtstd,  const float* __restrict__ p0d,
                const float* __restrict__ imuw, float* __restrict__ out, int B) {
  __shared__ float lds[WAVES * ELEM_F];
  __shared__ float qd[12];

  const int tid  = threadIdx.x;
  const int wave = tid >> 5;
  const int lane = tid & 31;
  const int lr   = lane & 15;
  const int lh   = lane >> 4;
  const int b    = blockIdx.x * WAVES + wave;
  const bool active = (b < B);
  float* W = lds + wave * ELEM_F;

  // IMU process-noise diagonal (shared by all batch elements)
  if (tid == 0) {
    const float nd[4] = {1e-7f, 1e-7f, 0.01f, 0.001f};
    for (int i = 0; i < 4; ++i) {
      float c = exp10f(4.0f * tanhf(imuw[i]));
      qd[i * 3 + 0] = nd[i] * c;
      qd[i * 3 + 1] = nd[i] * c;
      qd[i * 3 + 2] = nd[i] * c;
    }
  }

  // init: zero entire wave region (establishes the permanent zero padding of
  // rows/cols 18-19 in every matrix), then P diagonal + F constant blocks.
  if (active) {
    for (int idx = lane; idx < ELEM_F; idx += 32) W[idx] = 0.0f;
    if (lane < 18) {
      float d = p0d[lane];
      W[OFF_P + lane * LDA + lane] = d * d + 1e-12f;
    }
    if (lane < 3) {
      W[OFF_F + lane * LDA + 12 + lane]       = -1.0f;  // F[0:3,12:15] = -I
      W[OFF_F + (3 + lane) * LDA + 6 + lane]  =  1.0f;  // F[3:6, 6:9 ] =  I
      W[OFF_F + (6 + lane) * LDA + 9 + lane]  = -1.0f;  // F[6:9, 9:12] = -I
    }
  }
  __syncthreads();

  // per-lane process-noise weights for this half-wave's K slots
  float q0 = qd[2 * lh],     q1 = qd[2 * lh + 1];
  float q2 = qd[4 + 2 * lh], q3 = qd[4 + 2 * lh + 1];
  float q4 = qd[8 + 2 * lh], q5 = qd[8 + 2 * lh + 1];

  // ---------------- propagation: 11 steps ----------------
  for (int t = 0; t < 11; ++t) {
    float dt = 0.0f, w0 = 0, w1 = 0, w2 = 0, a0 = 0, a1 = 0, a2 = 0;
    if (active) {
      dt = dts[b * 11 + t];
      const float* wp = wa + b * 72 + t * 6;
      w0 = wp[0]; w1 = wp[1]; w2 = wp[2];
      a0 = wp[3]; a1 = wp[4]; a2 = wp[5];
      if (lane < 9) W[OFF_RS + lane] = Rg[b * 108 + t * 9 + lane];
    }
    __syncthreads();

    // F dynamic blocks (note: Rg_term + (accel - Rg_term) == accel)
    if (active && lane < 9) {
      int i = lane / 3, j = lane % 3;
      const float* Rm = W + OFF_RS;
      W[OFF_F + i * LDA + j] = -skew_ij(w0, w1, w2, i, j);   // -skew(w)
      float s = 0.0f;
      for (int m = 0; m < 3; ++m)
        s += Rm[i * 3 + m] * skew_ij(a0, a1, a2, m, j);
      W[OFF_F + (6 + i) * LDA + j]      = -s;                // -R@skew(accel)
      W[OFF_F + (6 + i) * LDA + 15 + j] = -Rm[i * 3 + j];    // -R
    }
    __syncthreads();

    // T1 = F @ F   (WMMA)
    if (active) matmul18(W + OFF_F, W + OFF_F, W + OFF_T1, lr, lh);
    __syncthreads();

    // Phi = I + F*dt + 0.5*F^2*dt^2  (interior 18x18 only; padding stays 0)
    if (active) {
      float hdt2 = 0.5f * dt * dt;
      for (int idx = lane; idx < 324; idx += 32) {
        int i = idx / 18, j = idx % 18;
        int o = i * LDA + j;
        W[OFF_PHI + o] = ((i == j) ? 1.0f : 0.0f)
                       + W[OFF_F + o] * dt + W[OFF_T1 + o] * hdt2;
      }
    }
    __syncthreads();

    // Phi[0:3,0:3] = exp_SO3(dt*w)^T
    if (active && lane < 9) {
      int i = lane / 3, j = lane % 3;
      float p0 = dt * w0, p1 = dt * w1, p2 = dt * w2;
      float th2 = p0 * p0 + p1 * p1 + p2 * p2;
      float th  = sqrtf(th2);
      bool  sm  = th < 1e-8f;
      float tt  = sm ? 1.0f : th;
      float Ac  = sm ? (1.0f - th2 / 6.0f)  : (sinf(tt) / tt);
      float Bc  = sm ? (0.5f - th2 / 24.0f) : ((1.0f - cosf(tt)) / (tt * tt));
      float k2 = 0.0f;
      for (int m = 0; m < 3; ++m)
        k2 += skew_ij(p0, p1, p2, j, m) * skew_ij(p0, p1, p2, m, i);
      W[OFF_PHI + i * LDA + j] = ((i == j) ? 1.0f : 0.0f)
                               + Ac * skew_ij(p0, p1, p2, j, i) + Bc * k2;
    }
    __syncthreads();

    // PhiG = Phi @ G  (G block-sparse), 18x12 stride 12
    if (active && lane < 18) {
      const float* Ph = W + OFF_PHI + lane * LDA;
      const float* Rm = W + OFF_RS;
      float* pg = W + OFF_PG + lane * 12;
      pg[0] = -Ph[0];  pg[1] = -Ph[1];  pg[2] = -Ph[2];     // -Phi[:,0:3]
      pg[3] =  Ph[12]; pg[4] =  Ph[13]; pg[5] =  Ph[14];    //  Phi[:,12:15]
      for (int c = 0; c < 3; ++c)                            // -Phi[:,6:9]@R
        pg[6 + c] = -(Ph[6] * Rm[c] + Ph[7] * Rm[3 + c] + Ph[8] * Rm[6 + c]);
      pg[9] = Ph[15]; pg[10] = Ph[16]; pg[11] = Ph[17];     //  Phi[:,15:18]
    }
    // T1 = Phi @ P   (WMMA) — independent of PhiG fill
    if (active) matmul18(W + OFF_PHI, W + OFF_P, W + OFF_T1, lr, lh);
    __syncthreads();

    // T2 = ((PhiG*q) @ PhiG^T) * dt  +  (Phi@P) @ Phi^T   (fused WMMA)
    if (active) {
      #pragma unroll
      for (int mi = 0; mi < 2; ++mi) {
        #pragma unroll
        for (int ni = 0; ni < 2; ++ni) {
          v8f acc = {};
          {
            v2f a, bv;
            a = fragA(W + OFF_PG, 12, mi, 0, lr, lh);
            a.x *= q0; a.y *= q1;
            bv = fragBT(W + OFF_PG, 12, ni, 0, lr, lh);
            acc = wmma4(a, bv, acc);
            a = fragA(W + OFF_PG, 12, mi, 4, lr, lh);
            a.x *= q2; a.y *= q3;
            bv = fragBT(W + OFF_PG, 12, ni, 4, lr, lh);
            acc = wmma4(a, bv, acc);
            a = fragA(W + OFF_PG, 12, mi, 8, lr, lh);
            a.x *= q4; a.y *= q5;
            bv = fragBT(W + OFF_PG, 12, ni, 8, lr, lh);
            acc = wmma4(a, bv, acc);
          }
          acc = acc * dt;
          #pragma unroll
          for (int k0 = 0; k0 < 20; k0 += 4) {
            v2f a  = fragA (W + OFF_T1,  LDA, mi, k0, lr, lh);
            v2f bv = fragBT(W + OFF_PHI, LDA, ni, k0, lr, lh);
            acc = wmma4(a, bv, acc);
          }
          storeD(W + OFF_T2, acc, mi, ni, lr, lh);
        }
      }
    }
    __syncthreads();

    // P = force_symmetrical(T2): upper triangle mirrored
    if (active) {
      for (int idx = lane; idx < 324; idx += 32) {
        int i = idx / 18, j = idx % 18;
        W[OFF_P + i * LDA + j] = (i <= j) ? W[OFF_T2 + i * LDA + j]
                                          : W[OFF_T2 + j * LDA + i];
      }
    }
    __syncthreads();
  }

  // ---------------- EKF update (scalar, small) ----------------
  float* HP = W + OFF_F;         // 6x18 (stride 18)
  float* S  = W + OFF_F + 108;   // 6x6
  float* SI = W + OFF_F + 144;   // 6x6
  float* yv = W + OFF_F + 180;   // 6
  float* es = W + OFF_F + 186;   // 18
  float* vc = W + OFF_F + 204;   // 6
  float* rr = W + OFF_F + 210;   // 6

  if (active && lane < 6) {
    float sd = (lane < 3) ? vrstd[b * 3 + lane] : vtstd[b * 3 + lane - 3];
    vc[lane] = (10.0f / (5.4f * 5.4f)) * exp10f(3.0f * tanhf(sd));
    rr[lane] = (lane < 3) ? (vrot[b * 3 + lane] - prot[b * 3 + lane])
                          : (vtrans[b * 3 + lane - 3] * 5.4f - ptrans[b * 3 + lane - 3]);
  }
  if (active && lane < 18) {
    int j = lane;
    const float* h0 = H0g + b * 9;
    const float* h1 = H1g + b * 9;
    const float* Pm = W + OFF_P;
    for (int i = 0; i < 3; ++i) {
      HP[i * 18 + j] = h0[i * 3 + 0] * Pm[0 * LDA + j] + h0[i * 3 + 1] * Pm[1 * LDA + j]
                     + h0[i * 3 + 2] * Pm[2 * LDA + j];
      HP[(3 + i) * 18 + j] = h1[i * 3 + 0] * Pm[0 * LDA + j] + h1[i * 3 + 1] * Pm[1 * LDA + j]
                           + h1[i * 3 + 2] * Pm[2 * LDA + j] + Pm[(3 + i) * LDA + j];
    }
  }
  __syncthreads();

  if (active && lane < 36) {
    int i = lane / 6, j = lane % 6;
    const float* hj = (j < 3) ? (H0g + b * 9 + j * 3) : (H1g + b * 9 + (j - 3) * 3);
    float s = HP[i * 18 + 0] * hj[0] + HP[i * 18 + 1] * hj[1] + HP[i * 18 + 2] * hj[2];
    if (j >= 3) s += HP[i * 18 + j];          // H[j][j]==1 for j in 3..5
    if (i == j) s += vc[i];                   // + vis_covar (diagonal)
    S[lane]  = s;
    SI[lane] = (i == j) ? 1.0f : 0.0f;
  }
  __syncthreads();

  if (active && lane == 0) {
    // 6x6 Gauss-Jordan inverse (S is SPD-dominated; no pivoting)
    for (int c = 0; c < 6; ++c) {
      float inv = 1.0f / S[c * 6 + c];
      for (int j = 0; j < 6; ++j) { S[c * 6 + j] *= inv; SI[c * 6 + j] *= inv; }
      for (int r2 = 0; r2 < 6; ++r2) {
        if (r2 == c) continue;
        float f = S[r2 * 6 + c];
        for (int j = 0; j < 6; ++j) {
          S[r2 * 6 + j]  -= f * S[c * 6 + j];
          SI[r2 * 6 + j] -= f * SI[c * 6 + j];
        }
      }
    }
    for (int c = 0; c < 6; ++c) {
      float s = 0.0f;
      for (int j = 0; j < 6; ++j) s += SI[c * 6 + j] * rr[j];
      yv[c] = s;                              // y = S^{-1} r
    }
  }
  __syncthreads();

  if (active && lane < 18) {                  // est = P H^T y = HP^T y (P sym)
    float s = 0.0f;
    for (int c = 0; c < 6; ++c) s += HP[c * 18 + lane] * yv[c];
    es[lane] = s;
  }
  __syncthreads();

  if (active && lane < 12) {
    float o;
    if (lane < 3) {
      const float* h0 = H0g + b * 9 + lane * 3;
      o = prot[b * 3 + lane] + h0[0] * es[0] + h0[1] * es[1] + h0[2] * es[2];
    } else if (lane < 6) {
      int r3 = lane - 3;
      const float* h1 = H1g + b * 9 + r3 * 3;
      o = ptrans[b * 3 + r3] + h1[0] * es[0] + h1[1] * es[1] + h1[2] * es[2] + es[3 + r3];
    } else if (lane < 9) {
      o = vel[b * 3 + lane - 6] + es[lane];
    } else {
      o = grav[b * 3 + lane - 9] + es[lane];
    }
    out[b * 12 + lane] = o;
  }
}

extern "C" void kernel_launch(void* const* d_in, const int* in_sizes, int n_in,
                              void* d_out, int out_size, void* d_ws, size_t ws_size,
                              hipStream_t stream) {
  const float* dts    = (const float*)d_in[0];
  const float* wa     = (const float*)d_in[1];
  const float* Rg     = (const float*)d_in[2];
  const float* vel    = (const float*)d_in[3];
  const float* grav   = (const float*)d_in[4];
  const float* H0g    = (const float*)d_in[5];
  const float* H1g    = (const float*)d_in[6];
  const float* prot   = (const float*)d_in[7];
  const float* ptrans = (const float*)d_in[8];
  const float* vrot   = (const float*)d_in[9];
  const float* vtrans = (const float*)d_in[10];
  const float* vrstd  = (const float*)d_in[11];
  const float* vtstd  = (const float*)d_in[12];
  const float* p0d    = (const float*)d_in[13];
  const float* imuw   = (const float*)d_in[14];

  int B = in_sizes[0] / 11;
  int blocks = (B + WAVES - 1) / WAVES;
  ekf_kernel<<<blocks, 32 * WAVES, 0, stream>>>(
      dts, wa, Rg, vel, grav, H0g, H1g, prot, ptrans, vrot, vtrans,
      vrstd, vtstd, p0d, imuw, (float*)d_out, B);
}